// RotationalQuantizer_33036888441546
// MI455X (gfx1250) — compile-verified
//
#include <hip/hip_runtime.h>
#include <math.h>

#define B_ROWS 2048
#define D_DIM  256
#define C_CODES 1024
#define EPSF   1e-6f
#define BETAF  0.25f
// v = ones/sqrt(256) -> every component is exactly 1/16
#define VCOMP  0.0625f

typedef __attribute__((ext_vector_type(2))) float v2f;
typedef __attribute__((ext_vector_type(8))) float v8f;

__device__ __forceinline__ float blockSum256(float v, float* red) {
    const int tid = threadIdx.x;
    red[tid] = v;
    __syncthreads();
    for (int off = 128; off > 0; off >>= 1) {
        if (tid < off) red[tid] += red[tid + off];
        __syncthreads();
    }
    float r = red[0];
    __syncthreads();
    return r;
}

// ---------------------------------------------------------------------------
// Kernel 1: x_can = R^T x  (rank-2 update, no DxD matrices)
// R^T x = x + u*(-vx + (d*vx - ux)/c) + v*(ux + (d*ux - vx)/c)
// ---------------------------------------------------------------------------
__global__ void __launch_bounds__(256)
rotquant_xcan(const float* __restrict__ x, const float* __restrict__ prev_q,
              float* __restrict__ xcan) {
    __shared__ float red[256];
    const int row = blockIdx.x;
    const int k = threadIdx.x;
    const float p  = prev_q[row * D_DIM + k];
    const float xv = x[row * D_DIM + k];
    const float spp = blockSum256(p * p, red);
    const float sp  = blockSum256(p, red);
    const float spx = blockSum256(p * xv, red);
    const float sx  = blockSum256(xv, red);
    const float nrm = fmaxf(sqrtf(spp), EPSF);
    const float inv = 1.0f / nrm;
    const float d  = (sp * inv) * VCOMP;      // u.v
    const float c  = 1.0f + d + EPSF;
    const float ux = spx * inv;               // u.x
    const float vx = sx * VCOMP;              // v.x
    const float alpha = -vx + (d * vx - ux) / c;
    const float beta  =  ux + (d * ux - vx) / c;
    xcan[row * D_DIM + k] = xv + alpha * (p * inv) + beta * VCOMP;
}

// ---------------------------------------------------------------------------
// Kernel 2: per-code squared norms
// ---------------------------------------------------------------------------
__global__ void __launch_bounds__(256)
rotquant_cnorm(const float* __restrict__ codes, float* __restrict__ cn) {
    __shared__ float red[256];
    const int j = blockIdx.x;
    const float cv = codes[j * D_DIM + threadIdx.x];
    const float s = blockSum256(cv * cv, red);
    if (threadIdx.x == 0) cn[j] = s;
}

// ---------------------------------------------------------------------------
// Kernel 3: WMMA distance GEMM + argmin.
// Block = 256 threads (8 wave32s) handles 16 rows x all 1024 codes.
// score_j = ||code_j||^2 - 2 * dot(x_can, code_j)   (argmin-equivalent)
// ---------------------------------------------------------------------------
__global__ void __launch_bounds__(256)
rotquant_argmin(const float* __restrict__ xcan, const float* __restrict__ codes,
                const float* __restrict__ cn, int* __restrict__ idxOut,
                float* __restrict__ idxF) {
    __shared__ float xs[16][258];     // padded stride: avoid 64-bank conflicts
    __shared__ float wbS[8][16];
    __shared__ int   wbI[8][16];

    const int tid = threadIdx.x;
    const int rowBase = blockIdx.x * 16;

    // Stage the 16x256 A tile in LDS
    for (int i = tid; i < 16 * D_DIM; i += 256) {
        const int r = i >> 8, k = i & 255;
        xs[r][k] = xcan[(rowBase + r) * D_DIM + k];
    }
    __syncthreads();

    const int wave = tid >> 5, lane = tid & 31;
    const int h = lane >> 4, r16 = lane & 15;

    float bs[8]; int bi[8];
#pragma unroll
    for (int r = 0; r < 8; ++r) { bs[r] = __builtin_inff(); bi[r] = 0; }

    // Each wave: 8 column tiles of 16 codes, K=256 in 64 WMMA steps of 4
    for (int t = 0; t < 8; ++t) {
        const int n = (wave * 8 + t) * 16 + r16;       // this lane's code column
        const float* cp = codes + (size_t)n * D_DIM;
        v8f acc = {0.f, 0.f, 0.f, 0.f, 0.f, 0.f, 0.f, 0.f};
#pragma unroll 8
        for (int s = 0; s < 64; ++s) {
            const int kb = s * 4 + h * 2;              // ISA: lanes 0-15 K={0,1}, 16-31 K={2,3}
            v2f a, b;
            a.x = xs[r16][kb]; a.y = xs[r16][kb + 1];  // A[m=r16][kb..kb+1]
            b.x = cp[kb];      b.y = cp[kb + 1];       // B[k][n] = codes[n][k]
            acc = __builtin_amdgcn_wmma_f32_16x16x4_f32(
                false, a, false, b, (short)0, acc, false, false);
        }
        const float cnv = cn[n];
#pragma unroll
        for (int r = 0; r < 8; ++r) {                  // acc[r]: m = r + 8*h, this lane's n
            const float sc = cnv - 2.0f * acc[r];
            if (sc < bs[r] || (sc == bs[r] && n < bi[r])) { bs[r] = sc; bi[r] = n; }
        }
    }

    // Reduce across the 16 lanes of each half (n varies across lanes)
#pragma unroll
    for (int off = 8; off >= 1; off >>= 1) {
#pragma unroll
        for (int r = 0; r < 8; ++r) {
            const float os = __shfl_xor(bs[r], off, 32);
            const int   oi = __shfl_xor(bi[r], off, 32);
            if (os < bs[r] || (os == bs[r] && oi < bi[r])) { bs[r] = os; bi[r] = oi; }
        }
    }
    if (r16 == 0) {
#pragma unroll
        for (int r = 0; r < 8; ++r) { wbS[wave][r + 8 * h] = bs[r]; wbI[wave][r + 8 * h] = bi[r]; }
    }
    __syncthreads();

    // Cross-wave reduction: wave w covered codes [w*128,(w+1)*128) -> ascending scan keeps
    // the reference's first-min tie rule.
    if (tid < 16) {
        float best = __builtin_inff(); int bidx = 0;
        for (int w = 0; w < 8; ++w) {
            const float sv = wbS[w][tid]; const int iv = wbI[w][tid];
            if (sv < best || (sv == best && iv < bidx)) { best = sv; bidx = iv; }
        }
        const int row = rowBase + tid;
        idxOut[row] = bidx;
        idxF[row] = (float)bidx;
    }
}

// ---------------------------------------------------------------------------
// Kernel 4: gather code, rotate back (quantized = R q_can), per-row sq error
// R q = q + u*(vq + (d*vq - uq)/c) + v*(-uq + (d*uq - vq)/c)
// ---------------------------------------------------------------------------
__global__ void __launch_bounds__(256)
rotquant_unrotate(const float* __restrict__ x, const float* __restrict__ prev_q,
                  const float* __restrict__ codes, const int* __restrict__ idx,
                  float* __restrict__ quantized, float* __restrict__ rowloss) {
    __shared__ float red[256];
    const int row = blockIdx.x;
    const int k = threadIdx.x;
    const float p = prev_q[row * D_DIM + k];
    const int j = idx[row];
    const float q = codes[(size_t)j * D_DIM + k];
    const float spp = blockSum256(p * p, red);
    const float sp  = blockSum256(p, red);
    const float spq = blockSum256(p * q, red);
    const float sq  = blockSum256(q, red);
    const float nrm = fmaxf(sqrtf(spp), EPSF);
    const float inv = 1.0f / nrm;
    const float d  = (sp * inv) * VCOMP;
    const float c  = 1.0f + d + EPSF;
    const float uq = spq * inv;
    const float vq = sq * VCOMP;
    const float gamma = vq + (d * vq - uq) / c;
    const float delta = -uq + (d * uq - vq) / c;
    const float qz = q + gamma * (p * inv) + delta * VCOMP;
    quantized[row * D_DIM + k] = qz;
    const float diff = x[row * D_DIM + k] - qz;
    const float sl = blockSum256(diff * diff, red);
    if (k == 0) rowloss[row] = sl;
}

// ---------------------------------------------------------------------------
// Kernel 5: loss = (1+beta) * mean(rowloss)   (deterministic single-block sum)
// ---------------------------------------------------------------------------
__global__ void __launch_bounds__(256)
rotquant_loss(const float* __restrict__ rowloss, float* __restrict__ lossOut) {
    __shared__ float red[256];
    float s = 0.f;
    for (int i = threadIdx.x; i < B_ROWS; i += 256) s += rowloss[i];
    const float t = blockSum256(s, red);
    if (threadIdx.x == 0) lossOut[0] = (1.0f + BETAF) * t / (float)B_ROWS;
}

extern "C" void kernel_launch(void* const* d_in, const int* in_sizes, int n_in,
                              void* d_out, int out_size, void* d_ws, size_t ws_size,
                              hipStream_t stream) {
    const float* x      = (const float*)d_in[0];
    const float* prev_q = (const float*)d_in[1];
    const float* codes  = (const float*)d_in[2];

    float* out       = (float*)d_out;
    float* quantized = out;                                  // B*D floats
    float* idxF      = out + (size_t)B_ROWS * D_DIM;         // B floats
    float* lossOut   = out + (size_t)B_ROWS * D_DIM + B_ROWS;// 1 float

    char* ws = (char*)d_ws;
    float* xcan    = (float*)ws;                                            // B*D f32
    float* cn      = (float*)(ws + (size_t)B_ROWS * D_DIM * 4);             // C f32
    int*   idx     = (int*)  (ws + (size_t)B_ROWS * D_DIM * 4 + C_CODES*4); // B i32
    float* rowloss = (float*)(ws + (size_t)B_ROWS * D_DIM * 4 + C_CODES*4 + B_ROWS*4);

    rotquant_xcan  <<<B_ROWS,      256, 0, stream>>>(x, prev_q, xcan);
    rotquant_cnorm <<<C_CODES,     256, 0, stream>>>(codes, cn);
    rotquant_argmin<<<B_ROWS / 16, 256, 0, stream>>>(xcan, codes, cn, idx, idxF);
    rotquant_unrotate<<<B_ROWS,    256, 0, stream>>>(x, prev_q, codes, idx, quantized, rowloss);
    rotquant_loss  <<<1,           256, 0, stream>>>(rowloss, lossOut);
}